// TextInput_35089882808402
// MI455X (gfx1250) — compile-verified
//
#include <hip/hip_runtime.h>

typedef __attribute__((ext_vector_type(2))) float v2f;
typedef __attribute__((ext_vector_type(8))) float v8f;

#define D_MODEL  1024
#define BATCH    4
#define SEQ      2048
#define T_TOT    (SEQ + 1)           // 2049 steps incl. BOS
#define NTILE    16                  // output columns per block
#define NBLOCKS  (D_MODEL / NTILE)   // 64 persistent blocks
#define NWAVES   4
#define NTHREADS (NWAVES * 32)       // 128 threads (4 wave32)
#define KCHUNK   (D_MODEL / NWAVES)  // 256 K per wave
#define NQUADS   (D_MODEL / 4)       // 256 K-quads total

// ---- dynamic LDS layout (floats); dynamic LDS region starts at offset 0 ----
// Interleaved A/B tile: per K-quad q (k = 4q..4q+3), a 128-float block:
//   [  0..31 ] W pair (k,k+1)   : n*2 + (k&1)
//   [ 32..63 ] W pair (k+2,k+3)
//   [ 64..95 ] h pair (k,k+1)   : m*2 + (k&1)   (16 padded M-rows)
//   [ 96..127] h pair (k+2,k+3)
// -> per WMMA: ONE ds_load_2addr_b64 (B at q*128+lb, A at q*128+64+lb),
//    both halves conflict-free across the 64 LDS banks.
#define AB_OFF   0
#define AB_SZ    (NQUADS * 128)             // 32768 fl = 128 KB
#define RED_OFF  (AB_OFF + AB_SZ)
#define RED_SZ   (NWAVES * 32 * 8)          //  1024 fl =   4 KB (partial C)
#define SMEM_FLOATS (AB_SZ + RED_SZ)

// W slot for element (n, k); h slot for element (m, k) is W slot + 64 (m<->n)
#define AB_W_IDX(n, k) \
  (((k) >> 2) * 128 + (((k) >> 1) & 1) * 32 + (n) * 2 + ((k) & 1))
#define AB_H_IDX(m, k) (AB_W_IDX(m, k) + 64)

// gfx1250 async global->LDS copy (per-lane scatter; tracked by ASYNCcnt).
__device__ __forceinline__ void async_load_b32_to_lds(unsigned int lds_byte_off,
                                                      const void* gaddr) {
  asm volatile("global_load_async_to_lds_b32 %0, %1, off"
               :
               : "v"(lds_byte_off), "v"(gaddr)
               : "memory");
}
__device__ __forceinline__ void wait_asynccnt0() {
  asm volatile("s_wait_asynccnt 0x0" ::: "memory");
}

__launch_bounds__(NTHREADS, 1)
__global__ void rnn_scan_kernel(const int* __restrict__ ids,
                                const float* __restrict__ emb,
                                const float* __restrict__ W,
                                float* __restrict__ out,
                                unsigned int* __restrict__ bar) {
  extern __shared__ float smem[];
  float* ab  = smem + AB_OFF;
  float* red = smem + RED_OFF;

  const int tid  = threadIdx.x;
  const int lane = tid & 31;
  const int wave = tid >> 5;
  const int n0   = blockIdx.x * NTILE;

  // ---- one-time: W^T slice into interleaved tile (coalesced along k) ----
  for (int i = tid; i < NTILE * D_MODEL; i += NTHREADS) {
    const int n = i >> 10;                 // 0..15
    const int k = i & (D_MODEL - 1);
    ab[AB_W_IDX(n, k)] = W[(size_t)(n0 + n) * D_MODEL + k];
  }
  // ---- one-time: zero all h slots (padded rows 4..15 stay zero forever;
  //      rows 0..3 are zero for the t=0 step where h_{-1}=0) ----
  for (int i = tid; i < NTILE * D_MODEL; i += NTHREADS) {
    ab[(i >> 6) * 128 + 64 + (i & 63)] = 0.0f;
  }
  __syncthreads();

  const int nl = lane & 15;                       // N / A-row index (ISA layout)
  const int lb = nl * 2 + ((lane >> 4) << 5);     // + 32 floats for K+2/K+3 half

  for (int t = 0; t < T_TOT; ++t) {
    // ---- stage h_{t-1} rows 0..3 (= out[:, t-1, :], L2-hot) into LDS via
    //      gfx1250 async scatter copies; swizzle folded into the LDS address.
    if (t > 0) {
#pragma unroll
      for (int j = 0; j < (BATCH * D_MODEL) / NTHREADS; ++j) {   // 32 per thread
        const int i = j * NTHREADS + tid;  // lanes -> consecutive k (coalesced)
        const int m = i >> 10;             // 0..3
        const int k = i & (D_MODEL - 1);
        const unsigned int ldsoff = (unsigned int)AB_H_IDX(m, k) * 4u;
        const float* g = &out[((size_t)m * T_TOT + (t - 1)) * D_MODEL + k];
        async_load_b32_to_lds(ldsoff, g);
      }
      wait_asynccnt0();
    }
    __syncthreads();

    // ---- partial GEMM: this wave's K chunk via V_WMMA_F32_16X16X4_F32.
    //      One ds_load_2addr_b64 per WMMA fetches both A and B float2s.
    v8f c = {};
    const int q0 = (wave * KCHUNK) >> 2;
#pragma unroll 8
    for (int q = 0; q < KCHUNK / 4; ++q) {
      const int off = (q0 + q) * 128 + lb;
      v2f b = *(const v2f*)&ab[off];        // W pair
      v2f a = *(const v2f*)&ab[off + 64];   // h pair
      c = __builtin_amdgcn_wmma_f32_16x16x4_f32(
          /*neg_a=*/false, a, /*neg_b=*/false, b,
          /*c_mod=*/(short)0, c, /*reuse_a=*/false, /*reuse_b=*/false);
    }

    *(v8f*)&red[(wave * 32 + lane) * 8] = c;
    __syncthreads();

    // ---- reduce 4 wave-partials, fuse embedding gather, write h_t ----
    if (wave == 0 && lane < NTILE) {
      // C/D layout: VGPR r on lanes 0-15 = row M=r -> batch rows are comps 0..3
#pragma unroll
      for (int m = 0; m < BATCH; ++m) {
        float acc = red[(0 * 32 + lane) * 8 + m] + red[(1 * 32 + lane) * 8 + m] +
                    red[(2 * 32 + lane) * 8 + m] + red[(3 * 32 + lane) * 8 + m];
        const int tok = (t == 0) ? 0 : ids[m * SEQ + (t - 1)];
        acc += emb[(size_t)tok * D_MODEL + (n0 + lane)];
        out[((size_t)m * T_TOT + t) * D_MODEL + (n0 + lane)] = acc;
      }
    } else if (wave == 1 && lane < BATCH && (t + 1) < T_TOT) {
      // gfx1250 global_prefetch of next step's embedding rows for this tile
      const int tok = ids[lane * SEQ + t];
      __builtin_prefetch(&emb[(size_t)tok * D_MODEL + n0], 0, 1);
    }

    // ---- device-wide barrier (monotonic counter, s_sleep backoff) ----
    __threadfence();
    __syncthreads();
    if (tid == 0) {
      __hip_atomic_fetch_add(bar, 1u, __ATOMIC_ACQ_REL, __HIP_MEMORY_SCOPE_AGENT);
      const unsigned int target = (unsigned int)(t + 1) * NBLOCKS;
      while (__hip_atomic_load(bar, __ATOMIC_ACQUIRE, __HIP_MEMORY_SCOPE_AGENT) <
             target) {
        __builtin_amdgcn_s_sleep(2);
      }
    }
    __syncthreads();
    __threadfence();  // acquire for all threads before reading out[t]
  }
}

extern "C" void kernel_launch(void* const* d_in, const int* in_sizes, int n_in,
                              void* d_out, int out_size, void* d_ws,
                              size_t ws_size, hipStream_t stream) {
  (void)in_sizes; (void)n_in; (void)out_size; (void)ws_size;
  const int*   ids = (const int*)d_in[0];    // (4, 2048) int32
  const float* emb = (const float*)d_in[1];  // (50257, 1024) f32
  const float* W   = (const float*)d_in[2];  // (1024, 1024) f32
  float* out = (float*)d_out;                // (4, 2049, 1024) f32
  unsigned int* bar = (unsigned int*)d_ws;

  // zero the barrier counter (capture-legal)
  hipMemsetAsync(d_ws, 0, 256, stream);

  const size_t shmem = SMEM_FLOATS * sizeof(float);  // 132 KB < 320 KB WGP LDS
  (void)hipFuncSetAttribute((const void*)rnn_scan_kernel,
                            hipFuncAttributeMaxDynamicSharedMemorySize,
                            (int)shmem);
  rnn_scan_kernel<<<dim3(NBLOCKS), dim3(NTHREADS), shmem, stream>>>(
      ids, emb, W, out, bar);
}